// WignerD_7232724927075
// MI455X (gfx1250) — compile-verified
//
#include <hip/hip_runtime.h>
#include <cmath>

typedef __attribute__((ext_vector_type(2))) float v2f;
typedef __attribute__((ext_vector_type(4))) float v4f;
typedef __attribute__((ext_vector_type(8))) float v8f;

#define DIMTOT 212
#define QPR 53                     // float4 quads per row (212/4)
#define LDSROWW 9
#define LDSPERB (DIMTOT*LDSROWW)   // 1908 floats per batch strip

// Routing of a 16-wide tile position to its irrep block: (l, n, A-base, local idx, valid)
struct Rt { int l, n, base, loc, valid; };

// Tile 1 = blockdiag(l=4 at [0,9), l=3 at [9,16))
__device__ __forceinline__ Rt route1(int p) {
  Rt r; int hi = (p >= 9);
  r.l    = hi ? 3 : 4;
  r.n    = 2*r.l + 1;
  r.base = hi ? 35 : 84;
  r.loc  = hi ? (p - 9) : p;
  r.valid = 1;
  return r;
}
// Tile 2 = blockdiag(l=0 at [0,1), l=1 at [1,4), l=2 at [4,9)); p>=9 invalid (K padding)
__device__ __forceinline__ Rt route2(int p) {
  Rt r; int ge1 = (p >= 1), ge4 = (p >= 4);
  r.l    = ge1 + ge4;
  r.n    = 2*r.l + 1;
  r.base = ge4 ? 10 : ge1;
  r.loc  = p - (ge4 ? 4 : ge1);
  r.valid = (p < 9);
  return r;
}

// L = Zalpha * A * Zbeta, element (row R, K index K). betaT: LDS {cos(m*b)}[m+4], {sin(m*b)}[9+m+4]
__device__ __forceinline__ float elemL(const float* __restrict__ As,
                                       const float* __restrict__ betaT,
                                       const Rt R, const Rt K,
                                       float ca, float sa) {
  int n = R.n, base = R.base;
  int i  = R.loc; i  = i  > n-1 ? n-1 : i;
  int kc = K.loc; kc = kc < 0 ? 0 : (kc > n-1 ? n-1 : kc);
  int mk = kc - R.l;                       // in [-4,4]
  float cb = betaT[mk + 4];
  float sb = betaT[9 + mk + 4];
  int ri = n-1-i, rk = n-1-kc;
  float w0 = As[base + i*n  + kc]*cb + As[base + i*n  + rk]*sb;
  float w1 = As[base + ri*n + kc]*cb + As[base + ri*n + rk]*sb;
  float v  = ca*w0 - sa*w1;
  int inb = R.valid & K.valid & (R.l == K.l);
  return inb ? v : 0.0f;
}
// R = A^T * Zgamma, element (K index K, col J). Gamma trig fixed per lane.
__device__ __forceinline__ float elemR(const float* __restrict__ As,
                                       const Rt K, const Rt J,
                                       float cg, float sg) {
  int n = J.n, base = J.base;
  int j  = J.loc; j  = j  > n-1 ? n-1 : j;
  int kc = K.loc; kc = kc < 0 ? 0 : (kc > n-1 ? n-1 : kc);
  int rj = n-1-j;
  float v = As[base + j*n + kc]*cg + As[base + rj*n + kc]*sg;
  int inb = K.valid & J.valid & (K.l == J.l);
  return inb ? v : 0.0f;
}

__global__ __launch_bounds__(256)
void wignerd_kernel(const float* __restrict__ alpha, const float* __restrict__ beta,
                    const float* __restrict__ gam, float* __restrict__ out,
                    const float* __restrict__ Ac)
{
  __shared__ float strip[2 * LDSPERB];
  __shared__ float As[165];
  __shared__ float betaT[2][18];

  const int tid  = threadIdx.x;
  const int wave = tid >> 5, lane = tid & 31;
  const int bb   = wave >> 2, sub = wave & 3;   // 2 batches/block, 4 waves/batch
  const int b    = blockIdx.x * 2 + bb;

  // Phase 0: zero the LDS block strips; stage A constants in LDS
  for (int t = tid; t < 2*LDSPERB; t += 256) strip[t] = 0.0f;
  if (tid < 165) As[tid] = Ac[tid];
  __syncthreads();

  // Phase 1: one wave per batch computes all D_l blocks via WMMA
  if (sub == 0) {
    float aB = beta[b];
    if (lane < 9) {
      float s_, c_;
      sincosf((float)(lane - 4) * aB, &s_, &c_);
      betaT[bb][lane]     = c_;
      betaT[bb][9 + lane] = s_;
    }
    const float aA = alpha[b], aG = gam[b];
    const int m = lane & 15, half = lane >> 4;
    const float* bt = betaT[bb];
    float* L = &strip[bb * LDSPERB];

    // ---- Tile 1: 16x16, K=16 ----
    {
      Rt Rm = route1(m);
      float ca, sa, cg, sg;
      sincosf((float)(Rm.loc - Rm.l) * aA, &sa, &ca);
      sincosf((float)(Rm.loc - Rm.l) * aG, &sg, &cg);
      v8f c1 = {0,0,0,0,0,0,0,0};
      #pragma unroll
      for (int s = 0; s < 4; ++s) {
        int k0 = 4*s + 2*half;
        Rt K0 = route1(k0), K1 = route1(k0 + 1);
        v2f av, bv;
        av.x = elemL(As, bt, Rm, K0, ca, sa);
        av.y = elemL(As, bt, Rm, K1, ca, sa);
        bv.x = elemR(As, K0, Rm, cg, sg);
        bv.y = elemR(As, K1, Rm, cg, sg);
        c1 = __builtin_amdgcn_wmma_f32_16x16x4_f32(false, av, false, bv, (short)0, c1, false, false);
      }
      // scatter: l=4 block (4 reps), l=3 block (8 reps)
      #pragma unroll
      for (int v = 0; v < 8; ++v) {
        int r = v + 8*half, col = m;
        float x1 = c1[v];
        if (r < 9 && col < 9) {
          #pragma unroll
          for (int k = 0; k < 4; ++k) L[(176 + 9*k + r)*LDSROWW + col] = x1;
        } else if (r >= 9 && col >= 9) {
          int rr = r-9, cc = col-9;
          #pragma unroll
          for (int k = 0; k < 8; ++k) L[(120 + 7*k + rr)*LDSROWW + cc] = x1;
        }
      }
    }
    // ---- Tile 2: 9x9, K padded to 12 ----
    {
      Rt Rm = route2(m);
      float ca, sa, cg, sg;
      sincosf((float)(Rm.loc - Rm.l) * aA, &sa, &ca);
      sincosf((float)(Rm.loc - Rm.l) * aG, &sg, &cg);
      v8f c2 = {0,0,0,0,0,0,0,0};
      #pragma unroll
      for (int s = 0; s < 3; ++s) {
        int k0 = 4*s + 2*half;
        Rt K0 = route2(k0), K1 = route2(k0 + 1);
        v2f av, bv;
        av.x = elemL(As, bt, Rm, K0, ca, sa);
        av.y = elemL(As, bt, Rm, K1, ca, sa);
        bv.x = elemR(As, K0, Rm, cg, sg);
        bv.y = elemR(As, K1, Rm, cg, sg);
        c2 = __builtin_amdgcn_wmma_f32_16x16x4_f32(false, av, false, bv, (short)0, c2, false, false);
      }
      // scatter: l=0 (32 reps), l=1 (16 reps), l=2 (8 reps)
      #pragma unroll
      for (int v = 0; v < 8; ++v) {
        int r = v + 8*half, col = m;
        float x2 = c2[v];
        if (r == 0 && col == 0) {
          #pragma unroll
          for (int k = 0; k < 32; ++k) L[k*LDSROWW] = x2;
        } else if (r >= 1 && r < 4 && col >= 1 && col < 4) {
          #pragma unroll
          for (int k = 0; k < 16; ++k) L[(32 + 3*k + r-1)*LDSROWW + (col-1)] = x2;
        } else if (r >= 4 && r < 9 && col >= 4 && col < 9) {
          #pragma unroll
          for (int k = 0; k < 8; ++k) L[(80 + 5*k + r-4)*LDSROWW + (col-4)] = x2;
        }
      }
    }
  }
  __syncthreads();

  // Phase 2: stream the whole 212x212 slab with b128 NT stores, splicing block values
  const float* src = &strip[bb * LDSPERB];
  v4f* outq = reinterpret_cast<v4f*>(out) + (size_t)b * (DIMTOT*QPR) + sub * (53*QPR);
  for (int q = lane; q < 53*QPR; q += 32) {
    int r  = q / QPR;
    int cq = q - r*QPR;
    int i  = sub*53 + r;
    // flat select-based row decode (unused divisions are harmless)
    int t1 = (i-32)/3, t2 = (i-80)/5, t3 = (i-120)/7, t4 = (i-176)/9;
    int off = i < 32 ? i
            : (i < 80 ? 32 + 3*t1
            : (i < 120 ? 80 + 5*t2
            : (i < 176 ? 120 + 7*t3 : 176 + 9*t4)));
    int n   = i < 32 ? 1 : (i < 80 ? 3 : (i < 120 ? 5 : (i < 176 ? 7 : 9)));
    v4f vst;
    int j0 = cq * 4;
    #pragma unroll
    for (int k = 0; k < 4; ++k) {
      int d  = j0 + k - off;
      int cl = d < 0 ? 0 : (d > 8 ? 8 : d);
      float x = src[i*LDSROWW + cl];
      vst[k] = ((unsigned)d < (unsigned)n) ? x : 0.0f;
    }
    __builtin_nontemporal_store(vst, &outq[q]);
  }
}

// ---- Host: real-SH Wigner matrices of _RX via Ivanic-Ruedenberg recursion ----
static void computeWigner(float* out165) {
  // l=1 matrix in basis (m=-1,0,1) ~ (y,z,x) for R = [[1,0,0],[0,0,1],[0,-1,0]]
  double R1[3][3] = {{0,1,0},{-1,0,0},{0,0,1}};
  static double Ml[5][81];
  Ml[0][0] = 1.0;
  for (int i = 0; i < 9; ++i) Ml[1][i] = R1[i/3][i%3];
  for (int l = 2; l <= 4; ++l) {
    int n = 2*l + 1, np = n - 2;
    const double* Mp = Ml[l-1];
    auto mp = [&](int a, int b)->double { return Mp[(a + l - 1)*np + (b + l - 1)]; };
    auto rr = [&](int i, int j)->double { return R1[i + 1][j + 1]; };
    auto P  = [&](int i, int a, int b)->double {
      if (b ==  l) return rr(i,1)*mp(a, l-1) - rr(i,-1)*mp(a, 1-l);
      if (b == -l) return rr(i,1)*mp(a, 1-l) + rr(i,-1)*mp(a, l-1);
      return rr(i,0)*mp(a, b);
    };
    for (int m1 = -l; m1 <= l; ++m1)
      for (int m2 = -l; m2 <= l; ++m2) {
        int am1 = m1 < 0 ? -m1 : m1, am2 = m2 < 0 ? -m2 : m2;
        double denom = (am2 == l) ? (double)(2*l)*(2*l - 1) : (double)(l + m2)*(l - m2);
        double u = sqrt((double)(l + m1)*(l - m1)/denom);
        double v = 0.5*sqrt((1.0 + (m1 == 0))*(l + am1 - 1.0)*(l + am1)/denom)*(1.0 - 2.0*(m1 == 0));
        double w = -0.5*sqrt((l - am1 - 1.0)*(l - am1)/denom)*(1.0 - (m1 == 0));
        double Uv = 0, Vv = 0, Wv = 0;
        if (u != 0.0) Uv = P(0, m1, m2);
        if (v != 0.0) {
          if (m1 == 0)     Vv = P(1, 1, m2) + P(-1, -1, m2);
          else if (m1 > 0) Vv = P(1, m1-1, m2)*sqrt(1.0 + (m1 == 1)) - P(-1, -m1+1, m2)*(1.0 - (m1 == 1));
          else             Vv = P(1, m1+1, m2)*(1.0 - (m1 == -1)) + P(-1, -m1-1, m2)*sqrt(1.0 + (m1 == -1));
        }
        if (w != 0.0) {
          if (m1 > 0)      Wv = P(1, m1+1, m2) + P(-1, -m1-1, m2);
          else if (m1 < 0) Wv = P(1, m1-1, m2) - P(-1, -m1+1, m2);
        }
        Ml[l][(m1 + l)*n + (m2 + l)] = u*Uv + v*Vv + w*Wv;
      }
  }
  const int offs[5] = {0, 1, 10, 35, 84};
  for (int l = 0; l <= 4; ++l) {
    int n = 2*l + 1;
    for (int i = 0; i < n*n; ++i) out165[offs[l] + i] = (float)Ml[l][i];
  }
}

extern "C" void kernel_launch(void* const* d_in, const int* in_sizes, int n_in,
                              void* d_out, int out_size, void* d_ws, size_t ws_size,
                              hipStream_t stream) {
  static float hA[165];
  computeWigner(hA);  // deterministic, recomputed every call; static buffer persists for graph replay
  hipMemcpyAsync(d_ws, hA, sizeof(hA), hipMemcpyHostToDevice, stream);

  const float* alpha = (const float*)d_in[0];
  const float* beta  = (const float*)d_in[1];
  const float* gam   = (const float*)d_in[2];
  int B = in_sizes[0];
  int grid = B / 2;  // 2 batch elements per 256-thread block
  wignerd_kernel<<<grid, 256, 0, stream>>>(alpha, beta, gam,
                                           (float*)d_out, (const float*)d_ws);
}